// GCN_429496729879
// MI455X (gfx1250) — compile-verified
//
#include <hip/hip_runtime.h>
#include <hip/hip_bf16.h>

typedef float v2f __attribute__((ext_vector_type(2)));
typedef float v8f __attribute__((ext_vector_type(8)));

// ---------------------------------------------------------------------------
// Degree / norm kernels
// ---------------------------------------------------------------------------
__global__ void init_deg_kernel(float* __restrict__ deg, int N) {
    int i = blockIdx.x * blockDim.x + threadIdx.x;
    if (i < N) deg[i] = 1.0f;  // self-loop contributes 1
}

__global__ void accum_deg_kernel(const int* __restrict__ col, float* __restrict__ deg, int E) {
    int e = blockIdx.x * blockDim.x + threadIdx.x;
    if (e < E) atomicAdd(&deg[col[e]], 1.0f);
}

__global__ void rsqrt_kernel(float* __restrict__ deg, int N) {
    int i = blockIdx.x * blockDim.x + threadIdx.x;
    if (i < N) deg[i] = rsqrtf(deg[i]);
}

// ---------------------------------------------------------------------------
// FP32 GEMM: C[M,Ncol] = A[M,K] @ B[K,Ncol] using V_WMMA_F32_16X16X4_F32.
// One wave32 computes one 16x16 output tile; K consumed 4 at a time.
// A-frag (16x4):  lane L -> row m = L&15, k-pair base kb = 2*(L>>4); vgpr0=k+kb, vgpr1=k+kb+1
// B-frag (4x16):  lane L -> col n = L&15, same kb mapping
// D (16x16):      lane L -> col n = L&15, vgpr r -> row 8*(L>>4)+r
// ---------------------------------------------------------------------------
__global__ __launch_bounds__(256)
void gemm_f32_wmma_kernel(const float* __restrict__ A, const float* __restrict__ B,
                          float* __restrict__ C, int M, int K, int Ncol) {
    const int wave  = (blockIdx.x * blockDim.x + threadIdx.x) >> 5;
    const int lane  = threadIdx.x & 31;
    const int tilesN = Ncol >> 4;
    const int tm = wave / tilesN;
    const int tn = wave - tm * tilesN;
    if (tm * 16 >= M) return;  // wave-uniform guard (EXEC stays all-ones for WMMA)

    const int half = lane >> 4;        // 0 or 1
    const int l15  = lane & 15;
    const int kb   = half * 2;         // k-pair base within each K=4 step

    const float* arow = A + (size_t)(tm * 16 + l15) * K;
    const int    ncol = tn * 16 + l15;

    v8f acc = {};
    #pragma unroll 4
    for (int k = 0; k < K; k += 4) {
        v2f a, b;
        a.x = arow[k + kb];
        a.y = arow[k + kb + 1];
        b.x = B[(size_t)(k + kb)     * Ncol + ncol];
        b.y = B[(size_t)(k + kb + 1) * Ncol + ncol];
        acc = __builtin_amdgcn_wmma_f32_16x16x4_f32(
            /*neg_a=*/false, a, /*neg_b=*/false, b,
            /*c_mod=*/(short)0, acc, /*reuse_a=*/false, /*reuse_b=*/false);
    }

    const int rbase = tm * 16 + half * 8;
    #pragma unroll
    for (int r = 0; r < 8; ++r) {
        C[(size_t)(rbase + r) * Ncol + ncol] = acc[r];
    }
}

// ---------------------------------------------------------------------------
// Aggregation: agg[i] = isq[i]^2 * H[i]  (self-loop term, optional bias)
// then edge scatter: agg[dst] += isq[src]*isq[dst] * H[src]
// ---------------------------------------------------------------------------
template<int CDIM, bool RELU_LATER>
__global__ void agg_init_kernel(const float* __restrict__ isq, const float* __restrict__ H,
                                const float* __restrict__ bias, float* __restrict__ agg, int N) {
    const int chunks = CDIM >> 2;
    int tid = blockIdx.x * blockDim.x + threadIdx.x;
    if (tid >= N * chunks) return;
    int i = tid / chunks;
    int c = (tid - i * chunks) << 2;
    float s = isq[i];
    float self = s * s;
    float4 v = *reinterpret_cast<const float4*>(H + (size_t)i * CDIM + c);
    float4 o;
    o.x = self * v.x; o.y = self * v.y; o.z = self * v.z; o.w = self * v.w;
    if (bias) {
        o.x += bias[c]; o.y += bias[c + 1]; o.z += bias[c + 2]; o.w += bias[c + 3];
    }
    *reinterpret_cast<float4*>(agg + (size_t)i * CDIM + c) = o;
}

template<int CDIM>
__global__ void scatter_kernel(const int* __restrict__ row, const int* __restrict__ col,
                               const float* __restrict__ isq, const float* __restrict__ H,
                               float* __restrict__ agg, int E) {
    const int chunks = CDIM >> 2;
    int tid = blockIdx.x * blockDim.x + threadIdx.x;
    if (tid >= E * chunks) return;
    int e = tid / chunks;
    int c = (tid - e * chunks) << 2;
    int s = row[e];
    int d = col[e];
    float nrm = isq[s] * isq[d];
    float4 v = *reinterpret_cast<const float4*>(H + (size_t)s * CDIM + c);
    float* dst = agg + (size_t)d * CDIM + c;
    atomicAdd(dst + 0, nrm * v.x);
    atomicAdd(dst + 1, nrm * v.y);
    atomicAdd(dst + 2, nrm * v.z);
    atomicAdd(dst + 3, nrm * v.w);
}

__global__ void bias_relu_kernel(const float* __restrict__ agg, const float* __restrict__ bias,
                                 float* __restrict__ h, int N, int CDIM) {
    int tid = blockIdx.x * blockDim.x + threadIdx.x;
    int total = N * (CDIM >> 2);
    if (tid >= total) return;
    int chunks = CDIM >> 2;
    int i = tid / chunks;
    int c = (tid - i * chunks) << 2;
    float4 v = *reinterpret_cast<const float4*>(agg + (size_t)i * CDIM + c);
    float4 o;
    o.x = fmaxf(v.x + bias[c], 0.0f);
    o.y = fmaxf(v.y + bias[c + 1], 0.0f);
    o.z = fmaxf(v.z + bias[c + 2], 0.0f);
    o.w = fmaxf(v.w + bias[c + 3], 0.0f);
    *reinterpret_cast<float4*>(h + (size_t)i * CDIM + c) = o;
}

// ---------------------------------------------------------------------------
// Launch
// ---------------------------------------------------------------------------
static inline int cdiv(long long a, int b) { return (int)((a + b - 1) / b); }

extern "C" void kernel_launch(void* const* d_in, const int* in_sizes, int n_in,
                              void* d_out, int out_size, void* d_ws, size_t ws_size,
                              hipStream_t stream) {
    const int IN_C = 128, HID_C = 128, OUT_C = 64;
    const float* x  = (const float*)d_in[0];
    const int*   ei = (const int*)d_in[1];
    const float* W1 = (const float*)d_in[2];
    const float* b1 = (const float*)d_in[3];
    const float* W2 = (const float*)d_in[4];
    const float* b2 = (const float*)d_in[5];
    float* out = (float*)d_out;

    const int N = in_sizes[0] / IN_C;   // 50000
    const int E = in_sizes[1] / 2;      // 800000
    const int* row = ei;        // edge_index[0] : source
    const int* col = ei + E;    // edge_index[1] : destination

    // Workspace layout (floats): isq[N] | bufA[N*128] | bufB[N*128]
    float* isq  = (float*)d_ws;
    float* bufA = isq + N;                  // H1, later reused for h (relu output)
    float* bufB = bufA + (size_t)N * HID_C; // agg1, later reused for H3

    const int T = 256;

    // 1) deg^{-1/2}
    init_deg_kernel<<<cdiv(N, T), T, 0, stream>>>(isq, N);
    accum_deg_kernel<<<cdiv(E, T), T, 0, stream>>>(col, isq, E);
    rsqrt_kernel<<<cdiv(N, T), T, 0, stream>>>(isq, N);

    // 2) H1 = x @ W1   [N,128]
    {
        long long waves = (long long)(N / 16) * (HID_C / 16);
        gemm_f32_wmma_kernel<<<cdiv(waves * 32, T), T, 0, stream>>>(x, W1, bufA, N, IN_C, HID_C);
    }

    // 3) agg1 = D^-1/2 A D^-1/2 H1   (self-loop init + edge scatter)
    agg_init_kernel<HID_C, true><<<cdiv((long long)N * (HID_C / 4), T), T, 0, stream>>>(
        isq, bufA, nullptr, bufB, N);
    scatter_kernel<HID_C><<<cdiv((long long)E * (HID_C / 4), T), T, 0, stream>>>(
        row, col, isq, bufA, bufB, E);

    // 4) h = relu(agg1 + b1)   (write into bufA, H1 fully consumed)
    bias_relu_kernel<<<cdiv((long long)N * (HID_C / 4), T), T, 0, stream>>>(
        bufB, b1, bufA, N, HID_C);

    // 5) H3 = h @ W2   [N,64]  (into bufB, agg1 fully consumed)
    {
        long long waves = (long long)(N / 16) * (OUT_C / 16);
        gemm_f32_wmma_kernel<<<cdiv(waves * 32, T), T, 0, stream>>>(bufA, W2, bufB, N, HID_C, OUT_C);
    }

    // 6) out = D^-1/2 A D^-1/2 H3 + b2
    agg_init_kernel<OUT_C, false><<<cdiv((long long)N * (OUT_C / 4), T), T, 0, stream>>>(
        isq, bufB, b2, out, N);
    scatter_kernel<OUT_C><<<cdiv((long long)E * (OUT_C / 4), T), T, 0, stream>>>(
        row, col, isq, bufB, out, E);
}